// DifferentiableBackprojection_5111011082346
// MI455X (gfx1250) — compile-verified
//
#include <hip/hip_runtime.h>
#include <math.h>

// ---- problem constants (match reference) ----
#define A_    120
#define V_    128
#define U_    128
#define NC    8
#define NZ_   96
#define NY_   96
#define NX_   96
#define DSO_  1000.0f
#define DSD_  1500.0f
#define DVOX_ 0.8f

// ---- tiling ----
#define TX 16   // x voxels per block
#define TY 4    // y voxels per block
#define TZ 16   // z voxels per block (4 per thread)
#define UT 32   // staged detector-u extent (footprint <= ~28)
#define VT 24   // staged detector-v extent (footprint <= ~22)
#define NTHR 256

#ifndef __has_builtin
#define __has_builtin(x) 0
#endif

#define AS1 __attribute__((address_space(1)))
#define AS3 __attribute__((address_space(3)))

typedef int v4i __attribute__((vector_size(16)));
typedef float f2v __attribute__((ext_vector_type(2)));

// 8-byte LDS load with only 4-byte alignment guarantee -> ds_load_2addr_b32 (read2)
__device__ __forceinline__ f2v lds_pair(const float* p) {
  f2v r;
  __builtin_memcpy(&r, p, sizeof(r));
  return r;
}

// ---- CDNA5 async global->LDS copy (ASYNCcnt path) ----
#if __has_builtin(__builtin_amdgcn_global_load_async_to_lds_b128)
__device__ __forceinline__ void async_copy_b128(const float* g, float* l) {
  typedef v4i AS1* gp_t;   // pointer to v4i in global address space
  typedef v4i AS3* lp_t;   // pointer to v4i in LDS address space (32-bit)
  // generic->AS1 is identity; generic shared addr low 32 bits == LDS offset
  gp_t gp = (gp_t)(unsigned long long)g;
  lp_t lp = (lp_t)(unsigned)(unsigned long long)l;
  __builtin_amdgcn_global_load_async_to_lds_b128(gp, lp, 0, 0);
}
#else
__device__ __forceinline__ void async_copy_b128(const float* g, float* l) {
  unsigned lo = (unsigned)(unsigned long long)l;      // LDS byte offset
  unsigned long long ga = (unsigned long long)g;      // 64-bit global address
  asm volatile("global_load_async_to_lds_b128 %0, %1, off"
               :: "v"(lo), "v"(ga) : "memory");
}
#endif

#if __has_builtin(__builtin_amdgcn_s_wait_asynccnt)
#define WAIT_ASYNC() __builtin_amdgcn_s_wait_asynccnt(0)
#else
#define WAIT_ASYNC() asm volatile("s_wait_asynccnt 0x0" ::: "memory")
#endif

// Exact tile footprint bounds from tile corners (iu is monotone per axis -> corner extrema).
__device__ __forceinline__ void tile_bounds(float c, float s,
                                            float cx0, float cx1,
                                            float cy0, float cy1,
                                            float cz0, float cz1,
                                            int& u_org, int& v_org) {
  float iumn = 1e30f, mgmn = 1e30f, mgmx = -1e30f;
  #pragma unroll
  for (int i = 0; i < 4; ++i) {
    float cx = (i & 1) ? cx1 : cx0;
    float cy = (i & 2) ? cy1 : cy0;
    float xr = cx * c + cy * s;
    float yr = -cx * s + cy * c;
    float mg = DSD_ / (DSO_ - xr);
    float u  = yr * mg + 0.5f * (U_ - 1);
    iumn = fminf(iumn, u);
    mgmn = fminf(mgmn, mg);
    mgmx = fmaxf(mgmx, mg);
  }
  float v00 = cz0 * mgmn, v01 = cz0 * mgmx, v10 = cz1 * mgmn, v11 = cz1 * mgmx;
  float ivmn = fminf(fminf(v00, v01), fminf(v10, v11)) + 0.5f * (V_ - 1);
  u_org = ((int)floorf(iumn)) & ~3;   // 16B-align u origin for b128 staging
  v_org = (int)floorf(ivmn);
}

// Stage one angle's footprint (8 channels x VT x UT) into LDS via async b128 loads.
// staged[ch][lv][lu] = det[ch][clamp(v_org+lv)][~clamp(u_org+lu)] -- exact for all
// in-range detector coords (u origin is 4-aligned, so the quad clamp only displaces
// coords > 127, which are only ever read with zero bilinear weight).
__device__ __forceinline__ void stage_angle(const float* __restrict__ sino,
                                            float (*dst)[VT][UT],
                                            int a, int u_org, int v_org, int tid) {
  const size_t angbase = (size_t)a * (V_ * U_);
  #pragma unroll
  for (int k = 0; k < (NC * VT * UT / 4) / NTHR; ++k) {   // 6 iterations
    int q   = tid + k * NTHR;            // quad index 0..1535
    int ch  = q / (VT * UT / 4);         // /192
    int r   = q - ch * (VT * UT / 4);
    int lv  = r >> 3;                    // 0..23
    int lu4 = (r & 7) << 2;              // 0,4,...,28
    int vg  = min(max(v_org + lv, 0), V_ - 1);
    int us  = min(max(u_org + lu4, 0), U_ - 4);
    const float* src = sino + (size_t)ch * (A_ * V_ * U_) + angbase + (size_t)vg * U_ + us;
    async_copy_b128(src, &dst[ch][lv][lu4]);
  }
}

__global__ __launch_bounds__(NTHR)
void DifferentiableBackprojection_kernel(const float* __restrict__ sino,
                                         const float* __restrict__ angles,
                                         float* __restrict__ out) {
  __shared__ __align__(16) float s_det[2][NC][VT][UT];  // 48 KB double buffer
  __shared__ float s_c[A_], s_s[A_];

  const int tid = threadIdx.x;
  if (tid < A_) {
    float cv, sv;
    __sincosf(angles[tid], &sv, &cv);   // native v_sin_f32/v_cos_f32 (range is [0,2pi))
    s_c[tid] = cv; s_s[tid] = sv;
  }
  __syncthreads();

  // block -> voxel tile
  const int b  = blockIdx.x;
  const int bx = b % (NX_ / TX);
  const int by = (b / (NX_ / TX)) % (NY_ / TY);
  const int bz = b / ((NX_ / TX) * (NY_ / TY));
  const int lx = tid & (TX - 1);
  const int ly = (tid >> 4) & (TY - 1);
  const int lz = tid >> 6;                   // 0..3
  const int gx = bx * TX + lx;
  const int gy = by * TY + ly;
  const float xm = (gx - 47.5f) * DVOX_;
  const float ym = (gy - 47.5f) * DVOX_;
  float zm[4]; int gz[4];
  #pragma unroll
  for (int j = 0; j < 4; ++j) {
    gz[j] = bz * TZ + lz + 4 * j;
    zm[j] = (gz[j] - 47.5f) * DVOX_;
  }

  // uniform tile corner coordinates (mm)
  const float cx0 = (bx * TX - 47.5f) * DVOX_;
  const float cx1 = (bx * TX + TX - 1 - 47.5f) * DVOX_;
  const float cy0 = (by * TY - 47.5f) * DVOX_;
  const float cy1 = (by * TY + TY - 1 - 47.5f) * DVOX_;
  const float cz0 = (bz * TZ - 47.5f) * DVOX_;
  const float cz1 = (bz * TZ + TZ - 1 - 47.5f) * DVOX_;

  float acc[4][NC];
  #pragma unroll
  for (int j = 0; j < 4; ++j)
    #pragma unroll
    for (int ch = 0; ch < NC; ++ch) acc[j][ch] = 0.0f;

  int uoA, voA, uoB, voB;   // per-buffer footprint origins
  tile_bounds(s_c[0], s_s[0], cx0, cx1, cy0, cy1, cz0, cz1, uoA, voA);
  stage_angle(sino, s_det[0], 0, uoA, voA, tid);

  int p = 0;
  for (int a = 0; a < A_; ++a) {
    WAIT_ASYNC();          // this wave's async loads into buf p are in LDS
    __syncthreads();       // all waves' loads done; everyone done reading buf p^1

    if (a + 1 < A_) {      // issue next angle into the other buffer (overlaps compute)
      int uo, vo;
      tile_bounds(s_c[a + 1], s_s[a + 1], cx0, cx1, cy0, cy1, cz0, cz1, uo, vo);
      if (p == 0) { uoB = uo; voB = vo; } else { uoA = uo; voA = vo; }
      stage_angle(sino, s_det[p ^ 1], a + 1, uo, vo, tid);
    }

    const int uo = p ? uoB : uoA;
    const int vo = p ? voB : voA;
    const float c = s_c[a], s = s_s[a];

    // geometry shared by this thread's 4 z-voxels (iu depends only on x,y)
    const float xr  = xm * c + ym * s;
    const float yr  = -xm * s + ym * c;
    const float mag = DSD_ / (DSO_ - xr);
    const float iu  = yr * mag + 0.5f * (U_ - 1);
    const float u0f = floorf(iu);
    const float fu  = iu - u0f;
    const int   u0  = min(max((int)u0f, 0), U_ - 1);
    const int   lu0 = min(max(u0 - uo, 0), UT - 2);   // taps at lu0, lu0+1
    const bool  vu  = (iu >= 0.0f) && (iu <= (float)(U_ - 1));
    const float w2  = mag * mag;

    #pragma unroll
    for (int j = 0; j < 4; ++j) {
      const float iv  = zm[j] * mag + 0.5f * (V_ - 1);
      const float v0f = floorf(iv);
      const float fv  = iv - v0f;
      const int   v0  = min(max((int)v0f, 0), V_ - 1);
      const int   lv0 = min(max(v0 - vo, 0), VT - 2); // taps at lv0, lv0+1
      const bool  valid = vu && (iv >= 0.0f) && (iv <= (float)(V_ - 1));
      const float w  = valid ? w2 : 0.0f;
      const float wa = (1.0f - fv) * (1.0f - fu) * w;
      const float wb = (1.0f - fv) * fu * w;
      const float wc = fv * (1.0f - fu) * w;
      const float wd = fv * fu * w;
      #pragma unroll
      for (int ch = 0; ch < NC; ++ch) {
        const float* row0 = &s_det[p][ch][lv0][lu0];
        const f2v r0 = lds_pair(row0);        // ds_load_2addr_b32 {s00,s01}
        const f2v r1 = lds_pair(row0 + UT);   // ds_load_2addr_b32 {s10,s11}
        acc[j][ch] += wa * r0.x + wb * r0.y + wc * r1.x + wd * r1.y;
      }
    }
    p ^= 1;
  }

  #pragma unroll
  for (int j = 0; j < 4; ++j)
    #pragma unroll
    for (int ch = 0; ch < NC; ++ch)
      out[(((size_t)ch * NZ_ + gz[j]) * NY_ + gy) * NX_ + gx] = acc[j][ch];
}

extern "C" void kernel_launch(void* const* d_in, const int* in_sizes, int n_in,
                              void* d_out, int out_size, void* d_ws, size_t ws_size,
                              hipStream_t stream) {
  const float* sino   = (const float*)d_in[0];   // [1,8,120,128,128] f32
  const float* angles = (const float*)d_in[1];   // [120] f32
  float* out = (float*)d_out;                    // [1,8,96,96,96] f32
  (void)in_sizes; (void)n_in; (void)out_size; (void)d_ws; (void)ws_size;
  dim3 grid((NX_ / TX) * (NY_ / TY) * (NZ_ / TZ));   // 864 blocks
  hipLaunchKernelGGL(DifferentiableBackprojection_kernel, grid, dim3(NTHR), 0, stream,
                     sino, angles, out);
}